// GCN_50672024158926
// MI455X (gfx1250) — compile-verified
//
#include <hip/hip_runtime.h>
#include <hip/hip_bf16.h>
#include <stdint.h>

typedef __attribute__((ext_vector_type(2))) float v2f;
typedef __attribute__((ext_vector_type(8))) float v8f;

#define LDSW 66   // padded LDS row stride (floats) to avoid bank conflicts

static __host__ int cdiv(long long a, long long b) { return (int)((a + b - 1) / b); }

// ---------------- degree / symmetric norm ----------------
__global__ void k_init_deg(float* deg, int n) {
  int i = blockIdx.x * blockDim.x + threadIdx.x;
  if (i < n) deg[i] = 1.0f;                 // self-loop (fill_value=1.0)
}

__global__ void k_deg_accum(const long long* __restrict__ col,
                            const float* __restrict__ w, float* deg, int e) {
  int i = blockIdx.x * blockDim.x + threadIdx.x;
  if (i < e) atomicAdd(&deg[(int)col[i]], w[i]);
}

__global__ void k_rsqrt(float* deg, int n) {
  int i = blockIdx.x * blockDim.x + threadIdx.x;
  if (i < n) { float d = deg[i]; deg[i] = (d > 0.f) ? rsqrtf(d) : 0.f; }
}

// ---------------- T = X @ W^T via V_WMMA_F32_16X16X4_F32 ----------------
// X: [n,64] row-major. W: [64(n_out),64(k)] row-major. T: [n,64].
// Block = 256 threads = 8 waves; each wave -> 16 output rows x 64 cols.
__global__ __launch_bounds__(256) void k_gemm64(const float* __restrict__ X,
                                                const float* __restrict__ W,
                                                float* __restrict__ T, int n) {
  __shared__ float Wl[64 * LDSW];     // W[nout][k] padded
  __shared__ float Xl[128 * LDSW];    // X tile padded
  const int tid  = threadIdx.x;
  const int row0 = blockIdx.x * 128;

  // Stage W (64x64 = 4096 floats), coalesced.
#pragma unroll
  for (int i = 0; i < 16; ++i) {
    int idx = tid + i * 256;
    Wl[(idx >> 6) * LDSW + (idx & 63)] = W[idx];
  }
  // Stage X tile (128x64 = 8192 floats), zero-padded past n.
#pragma unroll
  for (int i = 0; i < 32; ++i) {
    int idx = tid + i * 256;
    int r = idx >> 6, k = idx & 63;
    int gr = row0 + r;
    Xl[r * LDSW + k] = (gr < n) ? X[(size_t)gr * 64 + k] : 0.f;
  }
  __syncthreads();

  const int wave = tid >> 5, lane = tid & 31;
  const int half = lane >> 4, lm = lane & 15;
  const int lrow = wave * 16;
  if (row0 + lrow >= n) return;       // wave-uniform: EXEC stays all-ones below

  v8f acc0 = {}, acc1 = {}, acc2 = {}, acc3 = {};
#pragma unroll
  for (int k0 = 0; k0 < 64; k0 += 4) {
    // A frag (16x4 f32): lanes 0-15 -> K={k0,k0+1} of row M=lm; lanes 16-31 -> K={k0+2,k0+3}
    v2f a  = *(const v2f*)&Xl[(lrow + lm) * LDSW + k0 + 2 * half];
    // B frags (4x16 f32), B[k][nout] = W[nout][k]
    v2f b0 = *(const v2f*)&Wl[( 0 + lm) * LDSW + k0 + 2 * half];
    v2f b1 = *(const v2f*)&Wl[(16 + lm) * LDSW + k0 + 2 * half];
    v2f b2 = *(const v2f*)&Wl[(32 + lm) * LDSW + k0 + 2 * half];
    v2f b3 = *(const v2f*)&Wl[(48 + lm) * LDSW + k0 + 2 * half];
    acc0 = __builtin_amdgcn_wmma_f32_16x16x4_f32(false, a, false, b0, (short)0, acc0, false, false);
    acc1 = __builtin_amdgcn_wmma_f32_16x16x4_f32(false, a, false, b1, (short)0, acc1, false, false);
    acc2 = __builtin_amdgcn_wmma_f32_16x16x4_f32(false, a, false, b2, (short)0, acc2, false, false);
    acc3 = __builtin_amdgcn_wmma_f32_16x16x4_f32(false, a, false, b3, (short)0, acc3, false, false);
  }
  // C/D layout: vgpr r, lanes 0-15 -> M=r, lanes 16-31 -> M=r+8; N = lm + 16*tile
  const int orow = row0 + lrow + 8 * half;
#pragma unroll
  for (int r = 0; r < 8; ++r) {
    if (orow + r < n) {
      size_t base = (size_t)(orow + r) * 64 + lm;
      T[base +  0] = acc0[r];
      T[base + 16] = acc1[r];
      T[base + 32] = acc2[r];
      T[base + 48] = acc3[r];
    }
  }
}

// ---------------- zero ----------------
__global__ void k_zero(float* p, int n) {
  int i = blockIdx.x * blockDim.x + threadIdx.x;
  if (i < n) p[i] = 0.f;
}

// ---------------- edge scatter: agg[col] += norm * T[row]  (16 lanes/edge) ----
__global__ void k_scatter(const long long* __restrict__ row,
                          const long long* __restrict__ col,
                          const float* __restrict__ ew,
                          const float* __restrict__ dinv,
                          const float* __restrict__ T,
                          float* agg, int e) {
  int idx = blockIdx.x * blockDim.x + threadIdx.x;
  int eid = idx >> 4, q = idx & 15;
  if (eid >= e) return;
  int r = (int)row[eid], c = (int)col[eid];
  float nrm = dinv[r] * ew[eid] * dinv[c];
  float4 v = ((const float4*)(T + (size_t)r * 64))[q];
  float* dst = agg + (size_t)c * 64 + q * 4;
  atomicAdd(dst + 0, v.x * nrm);
  atomicAdd(dst + 1, v.y * nrm);
  atomicAdd(dst + 2, v.z * nrm);
  atomicAdd(dst + 3, v.w * nrm);
}

// -------- finalize: agg = [relu](agg + dinv^2 * T + b)   (self-loop folded in)
__global__ void k_finalize(float* agg, const float* __restrict__ T,
                           const float* __restrict__ dinv,
                           const float* __restrict__ b, int n, int do_relu) {
  int idx = blockIdx.x * blockDim.x + threadIdx.x;
  if (idx >= n * 16) return;
  int i = idx >> 4, q = idx & 15;
  float di = dinv[i]; float s = di * di;
  float4 a  = ((float4*)(agg + (size_t)i * 64))[q];
  float4 t  = ((const float4*)(T + (size_t)i * 64))[q];
  float4 bb = ((const float4*)b)[q];
  float4 o;
  o.x = a.x + s * t.x + bb.x;
  o.y = a.y + s * t.y + bb.y;
  o.z = a.z + s * t.z + bb.z;
  o.w = a.w + s * t.w + bb.w;
  if (do_relu) {
    o.x = fmaxf(o.x, 0.f); o.y = fmaxf(o.y, 0.f);
    o.z = fmaxf(o.z, 0.f); o.w = fmaxf(o.w, 0.f);
  }
  ((float4*)(agg + (size_t)i * 64))[q] = o;
}

// -------- pool: out[batch[i]] += h[i]; batch sorted -> running-sum + rare flush
__global__ void k_pool(const float* __restrict__ h,
                       const long long* __restrict__ batch,
                       float* out, int n) {
  int y = threadIdx.x >> 6, d = threadIdx.x & 63;
  int n0 = blockIdx.x * 256 + y * 64;
  if (n0 >= n) return;
  int n1 = n0 + 64; if (n1 > n) n1 = n;
  long long cur = batch[n0];
  float acc = 0.f;
  for (int i = n0; i < n1; ++i) {
    long long g = batch[i];
    if (g != cur) { atomicAdd(&out[(int)cur * 64 + d], acc); acc = 0.f; cur = g; }
    acc += h[(size_t)i * 64 + d];
  }
  atomicAdd(&out[(int)cur * 64 + d], acc);
}

// -------- head: logits = hG @ Wl^T + bl -> log_softmax ----------------
__global__ void k_head(const float* __restrict__ hG, const float* __restrict__ Wl,
                       const float* __restrict__ bl, float* out,
                       int n_graphs, int n_classes) {
  int g = blockIdx.x * blockDim.x + threadIdx.x;
  if (g >= n_graphs) return;
  float l[16];
  float m = -3.0e38f;
  for (int c = 0; c < n_classes; ++c) {
    float s = bl[c];
    for (int d = 0; d < 64; ++d) s += hG[g * 64 + d] * Wl[c * 64 + d];
    l[c] = s; m = fmaxf(m, s);
  }
  float sum = 0.f;
  for (int c = 0; c < n_classes; ++c) sum += __expf(l[c] - m);
  float lse = __logf(sum);
  for (int c = 0; c < n_classes; ++c) out[g * n_classes + c] = l[c] - m - lse;
}

extern "C" void kernel_launch(void* const* d_in, const int* in_sizes, int n_in,
                              void* d_out, int out_size, void* d_ws, size_t ws_size,
                              hipStream_t stream) {
  const float*     x    = (const float*)d_in[0];
  const long long* ei   = (const long long*)d_in[1];   // int64 [2, E]
  const long long* batch= (const long long*)d_in[2];   // int64 [N], sorted
  const float*     ew   = (const float*)d_in[3];
  const float*     W1   = (const float*)d_in[4];
  const float*     b1   = (const float*)d_in[5];
  const float*     W2   = (const float*)d_in[6];
  const float*     b2   = (const float*)d_in[7];
  const float*     W3   = (const float*)d_in[8];
  const float*     b3   = (const float*)d_in[9];
  const float*     Wl   = (const float*)d_in[10];
  const float*     bl   = (const float*)d_in[11];

  const int n = in_sizes[0] / 64;        // 100000
  const int e = in_sizes[3];             // 1000000
  const int n_classes = in_sizes[11];    // 10
  const int n_graphs  = out_size / (64 + n_classes); // 128

  const long long* row = ei;             // sources
  const long long* col = ei + e;         // targets

  float* dinv = (float*)d_ws;                  // [n]
  float* T    = dinv + n;                      // [n,64]
  float* agg  = T + (size_t)n * 64;            // [n,64], reused as activation
  float* out  = (float*)d_out;                 // hG at [0, n_graphs*64), logits after

  // symmetric normalization coefficients
  k_init_deg <<<cdiv(n, 256), 256, 0, stream>>>(dinv, n);
  k_deg_accum<<<cdiv(e, 256), 256, 0, stream>>>(col, ew, dinv, e);
  k_rsqrt    <<<cdiv(n, 256), 256, 0, stream>>>(dinv, n);

  const float* hin = x;
  const float* Ws[3] = {W1, W2, W3};
  const float* bs[3] = {b1, b2, b3};
  const int gemm_grid  = cdiv(n, 128);
  const int zero_grid  = cdiv((long long)n * 64, 256);
  const int scat_grid  = cdiv((long long)e * 16, 256);
  const int fin_grid   = cdiv((long long)n * 16, 256);
  for (int l = 0; l < 3; ++l) {
    k_gemm64  <<<gemm_grid, 256, 0, stream>>>(hin, Ws[l], T, n);
    k_zero    <<<zero_grid, 256, 0, stream>>>(agg, n * 64);
    k_scatter <<<scat_grid, 256, 0, stream>>>(row, col, ew, dinv, T, agg, e);
    k_finalize<<<fin_grid,  256, 0, stream>>>(agg, T, dinv, bs[l], n, l < 2);
    hin = agg;
  }

  k_zero<<<cdiv(out_size, 256), 256, 0, stream>>>(out, out_size);
  k_pool<<<cdiv(n, 256), 256, 0, stream>>>(agg, batch, out, n);
  k_head<<<cdiv(n_graphs, 128), 128, 0, stream>>>(out, Wl, bl,
                                                  out + (size_t)n_graphs * 64,
                                                  n_graphs, n_classes);
}